// PCENTransform_66460323938426
// MI455X (gfx1250) — compile-verified
//
#include <hip/hip_runtime.h>
#include <math.h>

// ---------------- problem constants (from reference) ----------------
#define B_    16
#define T_    16384
#define F_    128          // bins; one thread per bin, 128 threads/block (4 waves)
#define CHUNK 256          // frames per block
#define NC    (T_ / CHUNK) // 64 chunks per batch
#define TILE  32           // frames staged per LDS buffer (32*128*4 = 16 KB)
#define NTILES (CHUNK / TILE)

#define S_CONST     0.025f
#define A_CONST     (1.0f - S_CONST)   // 0.975
#define EPS_CONST   1e-6f
#define ALPHA_CONST 0.98f
#define DELTA_CONST 2.0f
#define SQRT_DELTA  1.41421356237f     // DELTA^0.5

typedef int v4i __attribute__((ext_vector_type(4)));
typedef __attribute__((address_space(1))) v4i* gv4_t;  // global int4*
typedef __attribute__((address_space(3))) v4i* lv4_t;  // LDS int4*

// ---------------- CDNA5 async global->LDS copy of one tile ----------------
// Tile is a contiguous TILE*F_*4 = 16384-byte region in global memory; LDS
// destination has the same flat layout. 128 threads x 16B x 8 iterations.
__device__ __forceinline__ void async_copy_tile(const float* __restrict__ gsrc,
                                                float* __restrict__ ldst) {
  const int tid = threadIdx.x;
#if defined(__gfx1250__) && __has_builtin(__builtin_amdgcn_global_load_async_to_lds_b128)
  gv4_t g = (gv4_t)gsrc + tid;  // 16-byte units
  lv4_t l = (lv4_t)ldst + tid;
#pragma unroll
  for (int j = 0; j < (TILE * F_ * 4) / (128 * 16); ++j) {
    __builtin_amdgcn_global_load_async_to_lds_b128(g + j * 128, l + j * 128,
                                                   /*offset=*/0, /*cpol=*/0);
  }
#else
  const float4* g4 = (const float4*)gsrc + tid;
  float4*       l4 = (float4*)ldst + tid;
#pragma unroll
  for (int j = 0; j < (TILE * F_) / (128 * 4); ++j) {
    __builtin_prefetch(g4 + j * 128 + (TILE * F_) / 4, 0, 1);
    l4[j * 128] = g4[j * 128];
  }
#endif
}

__device__ __forceinline__ void wait_async() {
#if __has_builtin(__builtin_amdgcn_s_wait_asynccnt)
  __builtin_amdgcn_s_wait_asynccnt(0);
#else
  asm volatile("s_wait_asynccnt 0" ::: "memory");
#endif
}

// ---------------- pass B: seed with exact carry-in, emit outputs ----------------
__global__ __launch_bounds__(F_) void pcen_passB(const float* __restrict__ x,
                                                 float* __restrict__ out,
                                                 const float* __restrict__ carry) {
  __shared__ float lds[2][TILE * F_];

  const int b = blockIdx.x / NC;
  const int c = blockIdx.x % NC;
  const int f = threadIdx.x;
  const bool first_chunk = (c == 0);

  const float* base  = x   + ((size_t)b * T_ + (size_t)c * CHUNK) * F_;
  float*       obase = out + ((size_t)b * T_ + (size_t)c * CHUNK) * F_;

  float m = first_chunk ? 0.0f : carry[((size_t)b * NC + (c - 1)) * F_ + f];

  async_copy_tile(base, lds[0]);

  for (int k = 0; k < NTILES; ++k) {
    wait_async();
    __syncthreads();
    if (k + 1 < NTILES) {
      async_copy_tile(base + (size_t)(k + 1) * TILE * F_, lds[(k + 1) & 1]);
    }
    const float* cur = lds[k & 1];
#pragma unroll
    for (int i = 0; i < TILE; ++i) {
      const float xv = cur[i * F_ + f];
      if (first_chunk && k == 0 && i == 0) {
        m = xv;  // reference: M[0] = x[0]
      } else {
        m = A_CONST * m + S_CONST * xv;
      }
      // (x * (M+eps)^(-alpha) + delta)^0.5 - delta^0.5
      const float inv_pow =
          __builtin_amdgcn_exp2f(-ALPHA_CONST * __builtin_amdgcn_logf(m + EPS_CONST));
      const float y = __builtin_amdgcn_sqrtf(xv * inv_pow + DELTA_CONST) - SQRT_DELTA;
      obase[(size_t)(k * TILE + i) * F_ + f] = y;
    }
    __syncthreads();
  }
}

// ---------------- pass A: chunk-end partial (recurrence from 0) ----------------
__global__ __launch_bounds__(F_) void pcen_passA(const float* __restrict__ x,
                                                 float* __restrict__ chunk_end) {
  __shared__ float lds[2][TILE * F_];

  const int b = blockIdx.x / NC;
  const int c = blockIdx.x % NC;
  const int f = threadIdx.x;
  const bool first_chunk = (c == 0);

  const float* base = x + ((size_t)b * T_ + (size_t)c * CHUNK) * F_;

  float m = 0.0f;  // chunk transfer term B_c accumulates from zero

  async_copy_tile(base, lds[0]);

  for (int k = 0; k < NTILES; ++k) {
    wait_async();
    __syncthreads();
    if (k + 1 < NTILES) {
      async_copy_tile(base + (size_t)(k + 1) * TILE * F_, lds[(k + 1) & 1]);
    }
    const float* cur = lds[k & 1];
#pragma unroll
    for (int i = 0; i < TILE; ++i) {
      const float xv = cur[i * F_ + f];
      if (first_chunk && k == 0 && i == 0) {
        m = xv;  // reference: M[0] = x[0]
      } else {
        m = A_CONST * m + S_CONST * xv;
      }
    }
    __syncthreads();
  }

  chunk_end[((size_t)b * NC + c) * F_ + f] = m;
}

// Small sequential scan over the 64 chunk summaries per (b, f) sequence.
__global__ __launch_bounds__(128) void pcen_scan(const float* __restrict__ chunk_end,
                                                 float* __restrict__ carry, float aq) {
  const int idx = blockIdx.x * blockDim.x + threadIdx.x;
  if (idx >= B_ * F_) return;
  const int b = idx / F_;
  const int f = idx % F_;
  float m = 0.0f;
  for (int c = 0; c < NC; ++c) {
    const float bc = chunk_end[((size_t)b * NC + c) * F_ + f];
    m = (c == 0) ? bc : (aq * m + bc);  // chunk 0 summary is already exact M-end
    carry[((size_t)b * NC + c) * F_ + f] = m;
  }
}

extern "C" void kernel_launch(void* const* d_in, const int* in_sizes, int n_in,
                              void* d_out, int out_size, void* d_ws, size_t ws_size,
                              hipStream_t stream) {
  (void)in_sizes; (void)n_in; (void)out_size; (void)ws_size;

  const float* x = (const float*)d_in[0];
  float* out = (float*)d_out;

  float* chunk_end = (float*)d_ws;                       // B*NC*F floats (512 KB)
  float* carry     = chunk_end + (size_t)B_ * NC * F_;   // B*NC*F floats (512 KB)

  const float aq = powf(A_CONST, (float)CHUNK);          // a^L, host-computed, deterministic

  pcen_passA<<<B_ * NC, F_, 0, stream>>>(x, chunk_end);
  pcen_scan<<<(B_ * F_ + 127) / 128, 128, 0, stream>>>(chunk_end, carry, aq);
  pcen_passB<<<B_ * NC, F_, 0, stream>>>(x, out, carry);
}